// DirectContractedVoxGO_1726576856422
// MI455X (gfx1250) — compile-verified
//
#include <hip/hip_runtime.h>

// Segmented exclusive prefix-scan (volume-rendering alpha compositing) for gfx1250.
// M = 8388608 samples, N = 16384 rays, ray_id sorted (contiguous segments).
//
// Phases (all L2-resident on MI455X's 192MB L2):
//   A) per-block (2048-elem tile) : alpha = -expm1(-softplus(d+shift)*itv),
//      l = -softplus(..)*itv ; block-local EXCLUSIVE scan of l via WMMA
//      tensor-core scan (16x16x4 f32), store e[i] + blockSums.
//   B) scan of blockSums -> blockOff (+ grand total).
//   C) at each ray boundary write segOff[r] = global exclusive cumsum at segment start.
//   D) weights[i] = exp((e[i]+blockOff) - segOff[ray]) * alpha[i].
//   E) alphainv_last[r] = exp(segOff[r+1]-segOff[r]).

typedef __attribute__((ext_vector_type(2))) float v2f;
typedef __attribute__((ext_vector_type(4))) float v4f;
typedef __attribute__((ext_vector_type(8))) float v8f;

#define TILE   2048   // elements per block (8 waves * 256)
#define WTILE  256    // elements per wave (one 16x16 f32 tile, column-major)
#define NWAVE  8

// padded LDS index: 2 floats of pad per 16-float row -> conflict-free WMMA gathers,
// keeps 8-byte alignment (pad is even).
__device__ __forceinline__ int pidx(int p) { return p + 2 * (p >> 4); }

__global__ __launch_bounds__(256) void k_alpha_scan(
    const float* __restrict__ density,
    const float* __restrict__ shift_p,
    const float* __restrict__ interval_p,
    float* __restrict__ alpha_out,     // d_out[0..M)
    float* __restrict__ e_out,         // ws: block-local exclusive scan of l
    float* __restrict__ blockSums)     // ws: per-block total of l
{
    const int lane = threadIdx.x & 31;
    const int wave = threadIdx.x >> 5;
    const int hf   = lane >> 4;        // lane half (A/B K-pair selector)
    const int n    = lane & 15;        // matrix column owned by this lane
    const float shift    = shift_p[0];
    const float interval = interval_p[0];
    const long long base = (long long)blockIdx.x * TILE + wave * WTILE;

    __shared__ float ltile[NWAVE][WTILE + 32];  // padded per-wave tile
    __shared__ float waveTot[NWAVE];

    // ---- coalesced load of 8 consecutive f32 per lane; fused math ----
    const float* gp = density + base + lane * 8;
    v4f d0 = *(const v4f*)(gp);
    v4f d1 = *(const v4f*)(gp + 4);
    float dv[8] = {d0.x, d0.y, d0.z, d0.w, d1.x, d1.y, d1.z, d1.w};
    float av[8];
    #pragma unroll
    for (int j = 0; j < 8; ++j) {
        float x  = dv[j] + shift;
        float sp = (x > 20.f) ? x : log1pf(expf(x));   // softplus
        float s  = sp * interval;
        av[j]    = -expm1f(-s);                        // alpha in [0,1)
        int   p  = lane * 8 + j;                       // tile position
        ltile[wave][pidx(p)] = -s;                     // l = log(1-alpha), exact
    }
    v4f a0 = {av[0], av[1], av[2], av[3]};
    v4f a1 = {av[4], av[5], av[6], av[7]};
    float* ap = alpha_out + base + lane * 8;
    *(v4f*)(ap)     = a0;
    *(v4f*)(ap + 4) = a1;
    __syncthreads();

    // ---- stage 1: Y = L * X  (per-column inclusive scans) ----
    // Tile element (k,n) = l[n*16 + k] (column-major). 4 chained K=4 WMMAs.
    // A 16x4 f32 layout (ISA 7.12.2): v0 = K{0|2}, v1 = K{1|3} by lane half.
    // B 4x16 f32 layout assumed mirror: v0 = row K{0|2}, v1 = row K{1|3}, N=lane&15.
    v8f acc = {0.f, 0.f, 0.f, 0.f, 0.f, 0.f, 0.f, 0.f};
    #pragma unroll
    for (int c = 0; c < 4; ++c) {
        const int k0 = 4 * c + 2 * hf;
        v2f a, b;
        a.x = (k0     <= n) ? 1.f : 0.f;   // L[m=n][k0]   (lower-tri incl diag)
        a.y = (k0 + 1 <= n) ? 1.f : 0.f;
        const int p = n * 16 + k0;         // X[k0][n]
        b.x = ltile[wave][pidx(p)];
        b.y = ltile[wave][pidx(p) + 1];    // X[k0+1][n] (pad preserves adjacency)
        acc = __builtin_amdgcn_wmma_f32_16x16x4_f32(
                  false, a, false, b, (short)0, acc, false, false);
    }

    // ---- stage 2: F = t_rep * S_strict + Y  (add cross-column offsets) ----
    // t[j] = Y[15][j] lives in acc[7], lanes 16..31.
    float y15 = acc[7];
    float tv[8];
    #pragma unroll
    for (int c = 0; c < 4; ++c) {
        tv[2 * c]     = __shfl(y15, 16 + 4 * c + 2 * hf,     32);
        tv[2 * c + 1] = __shfl(y15, 16 + 4 * c + 2 * hf + 1, 32);
    }
    #pragma unroll
    for (int c = 0; c < 4; ++c) {
        const int k0 = 4 * c + 2 * hf;
        v2f a, b;
        a.x = tv[2 * c];                   // t_rep[m][k] = t[k] for all m
        a.y = tv[2 * c + 1];
        b.x = (k0     < n) ? 1.f : 0.f;    // S_strict[k][n] = (k < n)
        b.y = (k0 + 1 < n) ? 1.f : 0.f;
        acc = __builtin_amdgcn_wmma_f32_16x16x4_f32(
                  false, a, false, b, (short)0, acc, false, false);
    }

    // ---- cross-wave combine ----
    float wtot = __shfl(acc[7], 31, 32);   // F[15][15] = wave inclusive total
    if (lane == 0) waveTot[wave] = wtot;
    __syncthreads();
    float woff = 0.f;
    for (int w = 0; w < wave; ++w) woff += waveTot[w];
    if (threadIdx.x == 0) {
        float s = 0.f;
        for (int w = 0; w < NWAVE; ++w) s += waveTot[w];
        blockSums[blockIdx.x] = s;
    }

    // ---- exclusive = inclusive - l + wave offset; lane owns 8 consecutive ----
    // C/D layout: vgpr j, lanes<16 -> (M=j,N=lane) = pos lane*16+j;
    //             lanes>=16 -> (M=j+8,N=lane-16) = pos (lane-16)*16+8+j.
    const int p0 = n * 16 + 8 * hf;
    float ev[8];
    #pragma unroll
    for (int j = 0; j < 8; ++j)
        ev[j] = acc[j] - ltile[wave][pidx(p0 + j)] + woff;
    v4f e0 = {ev[0], ev[1], ev[2], ev[3]};
    v4f e1 = {ev[4], ev[5], ev[6], ev[7]};
    float* ep = e_out + base + p0;
    *(v4f*)(ep)     = e0;
    *(v4f*)(ep + 4) = e1;
}

__global__ __launch_bounds__(256) void k_scan_blocksums(
    const float* __restrict__ blockSums, float* __restrict__ blockOff, int nB)
{
    __shared__ float ss[256];
    const int t = threadIdx.x;
    const int per = (nB + 255) / 256;
    const int s0 = t * per;
    float s = 0.f;
    for (int k = 0; k < per; ++k) { int i = s0 + k; if (i < nB) s += blockSums[i]; }
    ss[t] = s;
    __syncthreads();
    for (int d = 1; d < 256; d <<= 1) {           // Hillis-Steele inclusive
        float v = ss[t];
        float u = (t >= d) ? ss[t - d] : 0.f;
        __syncthreads();
        ss[t] = v + u;
        __syncthreads();
    }
    float run = (t > 0) ? ss[t - 1] : 0.f;        // exclusive
    for (int k = 0; k < per; ++k) {
        int i = s0 + k;
        if (i < nB) { blockOff[i] = run; run += blockSums[i]; }
    }
    if (t == 255) blockOff[nB] = ss[255];          // grand total
}

__global__ void k_segoff(const int* __restrict__ ray_id,
                         const float* __restrict__ e,
                         const float* __restrict__ blockOff,
                         float* __restrict__ segOff, int M_, int N_)
{
    const int i = blockIdx.x * blockDim.x + threadIdx.x;
    if (i >= M_) return;
    const int rid  = ray_id[i];
    const int prev = (i > 0) ? ray_id[i - 1] : -1;
    if (rid != prev) {
        float g = e[i] + blockOff[i / TILE];       // global exclusive cumsum at i
        for (int r = prev + 1; r <= rid; ++r) segOff[r] = g;  // covers empty rays
    }
    if (i == M_ - 1) {
        float tot = blockOff[M_ / TILE];
        for (int r = rid + 1; r <= N_; ++r) segOff[r] = tot;  // trailing + sentinel
    }
}

__global__ void k_weights(const int* __restrict__ ray_id,
                          const float* __restrict__ e,
                          const float* __restrict__ blockOff,
                          const float* __restrict__ segOff,
                          const float* __restrict__ alpha,
                          float* __restrict__ weights, int M_)
{
    const int i = blockIdx.x * blockDim.x + threadIdx.x;
    if (i >= M_) return;
    float g = e[i] + blockOff[i / TILE];
    float T = expf(g - segOff[ray_id[i]]);         // transmittance
    weights[i] = T * alpha[i];
}

__global__ void k_alphainv(const float* __restrict__ segOff,
                           float* __restrict__ out, int N_)
{
    const int r = blockIdx.x * blockDim.x + threadIdx.x;
    if (r < N_) out[r] = expf(segOff[r + 1] - segOff[r]);
}

extern "C" void kernel_launch(void* const* d_in, const int* in_sizes, int n_in,
                              void* d_out, int out_size, void* d_ws, size_t ws_size,
                              hipStream_t stream)
{
    const float* density  = (const float*)d_in[0];
    const int*   ray_id   = (const int*)d_in[1];
    const float* shift    = (const float*)d_in[2];
    const float* interval = (const float*)d_in[3];
    const int M_ = in_sizes[0];                  // 8388608 (multiple of TILE)
    const int N_ = out_size - 2 * M_;            // 16384 rays
    const int nB = M_ / TILE;

    float* alpha    = (float*)d_out;
    float* weights  = alpha + M_;
    float* alphainv = weights + M_;

    float* e         = (float*)d_ws;             // M floats
    float* blockSums = e + M_;                   // nB floats
    float* blockOff  = blockSums + nB;           // nB+1 floats
    float* segOff    = blockOff + (nB + 1);      // N+1 floats

    k_alpha_scan<<<nB, 256, 0, stream>>>(density, shift, interval,
                                         alpha, e, blockSums);
    k_scan_blocksums<<<1, 256, 0, stream>>>(blockSums, blockOff, nB);
    k_segoff<<<(M_ + 255) / 256, 256, 0, stream>>>(ray_id, e, blockOff, segOff, M_, N_);
    k_weights<<<(M_ + 255) / 256, 256, 0, stream>>>(ray_id, e, blockOff, segOff,
                                                    alpha, weights, M_);
    k_alphainv<<<(N_ + 255) / 256, 256, 0, stream>>>(segOff, alphainv, N_);
}